// QNetwork_40424232190269
// MI455X (gfx1250) — compile-verified
//
#include <hip/hip_runtime.h>

typedef __attribute__((ext_vector_type(2))) float v2f;
typedef __attribute__((ext_vector_type(8))) float v8f;
typedef __attribute__((ext_vector_type(4))) unsigned int u32x4;
typedef __attribute__((ext_vector_type(8))) int i32x8;
typedef __attribute__((ext_vector_type(4))) int i32x4;

#define BDIM 256

// Problem sizes (fixed by the reference)
#define BSZ   4
#define ASZ   8
#define NB    64
#define NC    32
#define DBOXD 4096
#define DPOSD 4
#define DMAPD 40
#define DCATD 4136              // DBOX + DMAP
#define DIN2D 8272              // 2*DCAT
#define H2D   1024
#define MD    (BSZ*ASZ*NB)      // 2048 rows for d-scores
#define MS    (BSZ*ASZ*NC)      // 1024 rows for s-scores
#define MTOT  (MD+MS)           // 3072
#define NBA   (BSZ*ASZ)         // 32

#if defined(__HIP_DEVICE_COMPILE__) && __has_builtin(__builtin_amdgcn_tensor_load_to_lds) && __has_builtin(__builtin_amdgcn_s_wait_tensorcnt)
#define USE_TDM 1
#else
#define USE_TDM 0
#endif

#if USE_TDM
// LDS byte offset of a __shared__ object (generic -> AS3 -> int)
__device__ __forceinline__ unsigned lds_off(const void* p) {
    return (unsigned)(unsigned long long)(__attribute__((address_space(3))) const void*)p;
}
#endif

// ---------------------------------------------------------------------------
// Kernel A: out[r, coff+0:4096] = box[r,:]; out[r, coff+4096:4136] = pos[r] @ W1^T
// ---------------------------------------------------------------------------
__global__ __launch_bounds__(BDIM)
void cat_rows(const float* __restrict__ box, const float* __restrict__ pos,
              const float* __restrict__ W1, float* __restrict__ out,
              int ldo, int coff)
{
    const int r   = blockIdx.x;
    const int tid = threadIdx.x;
    __shared__ float p[DPOSD];
    if (tid < DPOSD) p[tid] = pos[(size_t)r * DPOSD + tid];
    __syncthreads();

    float* orow = out + (size_t)r * ldo + coff;
    const float* brow = box + (size_t)r * DBOXD;
    for (int c = tid * 4; c < DBOXD; c += BDIM * 4)
        *(float4*)(orow + c) = *(const float4*)(brow + c);
    if (tid < DMAPD) {
        float acc = W1[tid*4+0]*p[0] + W1[tid*4+1]*p[1]
                  + W1[tid*4+2]*p[2] + W1[tid*4+3]*p[3];
        orow[DBOXD + tid] = acc;
    }
}

// ---------------------------------------------------------------------------
// Kernel B: WMMA f32 GEMM:  Out[M,N] = act( X[M,K] @ W[N,K]^T + bias )
// Block = 256 threads (8 waves), M-tile 32, N-tile 256. Each wave owns a
// 32x32 patch of D as a 2x2 grid of v_wmma_f32_16x16x4_f32 accumulators
// (A and B fragments each reused twice). K staged through double-buffered
// LDS in chunks of 16; W tiles DMA'd by the Tensor Data Mover when in-bounds.
// ---------------------------------------------------------------------------
#define MT   32
#define KC   16
#define WPAD 20   // padded LDS row (floats): keeps 16B alignment, breaks bank aliasing

__global__ __launch_bounds__(BDIM)
void gemm_wmma(const float* __restrict__ X, int ldx,
               const float* __restrict__ W, int ldw,
               const float* __restrict__ bias, int act,
               float* __restrict__ Out, int ldo,
               int M, int N, int K)
{
    __shared__ float Xs[2][MT][WPAD];
    __shared__ float Ws[2][256][WPAD];

    const int m0   = blockIdx.x * MT;
    const int n0   = blockIdx.y * 256;
    const int tid  = threadIdx.x;
    const int wave = tid >> 5;
    const int lane = tid & 31;
    const int lrow = lane & 15;       // m (A) / n (B) within a 16-wide tile
    const int kh   = lane >> 4;       // which k-pair this half-wave holds
    const int nt   = (K + KC - 1) / KC;

    v8f acc00 = {};   // (m-tile 0, n-tile 0)
    v8f acc01 = {};   // (m-tile 0, n-tile 1)
    v8f acc10 = {};   // (m-tile 1, n-tile 0)
    v8f acc11 = {};   // (m-tile 1, n-tile 1)

    auto stageX = [&](int buf, int kb) {
        if (tid < MT * 4) {
            int r = tid >> 2, c0 = (tid & 3) * 4;
            int gm = m0 + r;
            float4 v = make_float4(0.f, 0.f, 0.f, 0.f);
            if (gm < M) {
                const float* xr = X + (size_t)gm * ldx + kb;
                if (kb + KC <= K) {
                    v = *(const float4*)(xr + c0);
                } else {
                    float t0 = (kb + c0 + 0 < K) ? xr[c0 + 0] : 0.f;
                    float t1 = (kb + c0 + 1 < K) ? xr[c0 + 1] : 0.f;
                    float t2 = (kb + c0 + 2 < K) ? xr[c0 + 2] : 0.f;
                    float t3 = (kb + c0 + 3 < K) ? xr[c0 + 3] : 0.f;
                    v = make_float4(t0, t1, t2, t3);
                }
            }
            *(float4*)&Xs[buf][r][c0] = v;
        }
    };

    auto stageW = [&](int buf, int kb) {
        const bool full = (n0 + 256 <= N) && (kb + KC <= K);
#if USE_TDM
        if (full) {
            if (wave == 0) {
                // Tensor DMA: 2D tile, 16 x 256 elements of 4B, row stride ldw,
                // LDS padded 16 DW data + 4 DW pad per row (matches WPAD=20).
                unsigned long long ga = (unsigned long long)(const void*)W
                                      + ((size_t)n0 * (size_t)ldw + (size_t)kb) * 4ull;
                u32x4 g0;
                g0[0] = 1u;                                   // count=1, user mode
                g0[1] = lds_off(&Ws[buf][0][0]);              // lds_addr (bytes)
                g0[2] = (unsigned)ga;                         // global_addr[31:0]
                g0[3] = (unsigned)(ga >> 32) | (2u << 30);    // addr[56:32] | type=2
                i32x8 g1;
                g1[0] = (2 << 16)      // data_size = 4B
                      | (1 << 20)      // pad_enable
                      | (3 << 22)      // pad_interval: 16 DWORDs
                      | (3 << 25);     // pad_amount:   4 DWORDs
                g1[1] = (int)(16u << 16);    // tensor_dim0 = 16   (bits 79:48)
                g1[2] = (int)(256u << 16);   // tensor_dim1 = 256  (bits 111:80)
                g1[3] = (int)(16u << 16);    // tile_dim0 = 16     (bits 127:112)
                g1[4] = 256;                 // tile_dim1 = 256    (bits 143:128)
                g1[5] = ldw;                 // tensor_dim0_stride (bits 207:160)
                g1[6] = 0;
                g1[7] = 0;
                i32x4 g2 = {0, 0, 0, 0};     // group2 (unused, 2D tensor)
                i32x4 g3 = {0, 0, 0, 0};     // group3 (unused, 2D tensor)
                i32x8 g4 = {0, 0, 0, 0, 0, 0, 0, 0};  // extra group (unused)
                __builtin_amdgcn_tensor_load_to_lds(g0, g1, g2, g3, g4, 0);
            }
            return;
        }
#endif
        // Manual path (fallback + N/K edge tiles): one row per thread, b128 ops.
        int gn = n0 + tid;
        float4 v0 = make_float4(0.f,0.f,0.f,0.f), v1 = v0, v2 = v0, v3 = v0;
        if (gn < N) {
            const float* wr = W + (size_t)gn * ldw + kb;
            if (full || kb + KC <= K) {
                v0 = ((const float4*)wr)[0];
                v1 = ((const float4*)wr)[1];
                v2 = ((const float4*)wr)[2];
                v3 = ((const float4*)wr)[3];
            } else {
                float t[KC];
                for (int j = 0; j < KC; ++j) t[j] = (kb + j < K) ? wr[j] : 0.f;
                v0 = make_float4(t[0], t[1], t[2],  t[3]);
                v1 = make_float4(t[4], t[5], t[6],  t[7]);
                v2 = make_float4(t[8], t[9], t[10], t[11]);
                v3 = make_float4(t[12],t[13],t[14], t[15]);
            }
        }
        float* dst = &Ws[buf][tid][0];
        ((float4*)dst)[0] = v0;
        ((float4*)dst)[1] = v1;
        ((float4*)dst)[2] = v2;
        ((float4*)dst)[3] = v3;
    };

    auto compute = [&](int buf) {
        const int nr = wave * 32 + lrow;
#pragma unroll
        for (int ks = 0; ks < KC; ks += 4) {
            v2f a0, a1, b0, b1;
            a0.x = Xs[buf][lrow     ][ks + 2*kh + 0];
            a0.y = Xs[buf][lrow     ][ks + 2*kh + 1];
            a1.x = Xs[buf][lrow + 16][ks + 2*kh + 0];
            a1.y = Xs[buf][lrow + 16][ks + 2*kh + 1];
            b0.x = Ws[buf][nr     ][ks + 2*kh + 0];
            b0.y = Ws[buf][nr     ][ks + 2*kh + 1];
            b1.x = Ws[buf][nr + 16][ks + 2*kh + 0];
            b1.y = Ws[buf][nr + 16][ks + 2*kh + 1];
            acc00 = __builtin_amdgcn_wmma_f32_16x16x4_f32(
                        false, a0, false, b0, (short)0, acc00, false, false);
            acc01 = __builtin_amdgcn_wmma_f32_16x16x4_f32(
                        false, a0, false, b1, (short)0, acc01, false, false);
            acc10 = __builtin_amdgcn_wmma_f32_16x16x4_f32(
                        false, a1, false, b0, (short)0, acc10, false, false);
            acc11 = __builtin_amdgcn_wmma_f32_16x16x4_f32(
                        false, a1, false, b1, (short)0, acc11, false, false);
        }
    };

    stageX(0, 0);
    stageW(0, 0);
    for (int t = 0; t < nt; ++t) {
#if USE_TDM
        if (wave == 0) __builtin_amdgcn_s_wait_tensorcnt(0);
#endif
        __syncthreads();
        if (t + 1 < nt) {               // stage next tile into the other buffer
            stageX((t + 1) & 1, (t + 1) * KC);
            stageW((t + 1) & 1, (t + 1) * KC);
        }
        compute(t & 1);
    }

    // Epilogue. D layout: VGPR v, lanes 0-15 -> M=v, N=lane; 16-31 -> M=v+8.
    const int gnA = n0 + wave * 32 + lrow;
    const int gnB = gnA + 16;
#pragma unroll
    for (int v = 0; v < 8; ++v) {
        int gm0 = m0 + v + 8 * kh;        // m-tile 0
        int gm1 = gm0 + 16;               // m-tile 1
        if (gm0 < M) {
            if (gnA < N) {
                float val = acc00[v] + (bias ? bias[gnA] : 0.0f);
                if (act) val = tanhf(val);
                Out[(size_t)gm0 * ldo + gnA] = val;
            }
            if (gnB < N) {
                float val = acc01[v] + (bias ? bias[gnB] : 0.0f);
                if (act) val = tanhf(val);
                Out[(size_t)gm0 * ldo + gnB] = val;
            }
        }
        if (gm1 < M) {
            if (gnA < N) {
                float val = acc10[v] + (bias ? bias[gnA] : 0.0f);
                if (act) val = tanhf(val);
                Out[(size_t)gm1 * ldo + gnA] = val;
            }
            if (gnB < N) {
                float val = acc11[v] + (bias ? bias[gnB] : 0.0f);
                if (act) val = tanhf(val);
                Out[(size_t)gm1 * ldo + gnB] = val;
            }
        }
    }
}

// ---------------------------------------------------------------------------
// Kernel C: score[r] = exp( H[r,:] . W3 )   (one block per row)
// ---------------------------------------------------------------------------
__global__ __launch_bounds__(BDIM)
void score_kernel(const float* __restrict__ H, const float* __restrict__ W3,
                  float* __restrict__ score)
{
    const int r = blockIdx.x, tid = threadIdx.x;
    __shared__ float red[BDIM];
    float acc = 0.0f;
    for (int n = tid; n < H2D; n += BDIM) acc += H[(size_t)r * H2D + n] * W3[n];
    red[tid] = acc;
    __syncthreads();
    for (int s = BDIM / 2; s > 0; s >>= 1) {
        if (tid < s) red[tid] += red[tid + s];
        __syncthreads();
    }
    if (tid == 0) score[r] = expf(red[0]);
}

// ---------------------------------------------------------------------------
// Kernel D: ragged-masked normalization + attention pooling.
// Blocks 0..31  : f[ba]  = sum_n (d[ba,n]/zd) * allb[ba,n,:]   (allb = X rows)
// Blocks 32..63 : sv[ba] = sum_n (s[ba,n]/zs) * chb[ba,n,:]    (zs guard for 0)
// ---------------------------------------------------------------------------
__global__ __launch_bounds__(BDIM)
void pool_kernel(const float* __restrict__ score_d, const float* __restrict__ score_s,
                 const float* __restrict__ X,
                 const int* __restrict__ box_num, const int* __restrict__ chosen_num,
                 float* __restrict__ f, float* __restrict__ sv)
{
    const bool is_s = (blockIdx.x >= NBA);
    const int  ba   = blockIdx.x & (NBA - 1);
    const int  b    = ba / ASZ;
    const int  nit  = is_s ? NC : NB;
    const int  lim  = is_s ? chosen_num[b] : box_num[b];
    const float* sc = is_s ? (score_s + (size_t)ba * NC) : (score_d + (size_t)ba * NB);
    const int  tid  = threadIdx.x;

    __shared__ float red[BDIM];
    __shared__ float w[NB];

    float acc = 0.0f;
    if (tid < nit && tid < lim) acc = sc[tid];
    red[tid] = acc;
    __syncthreads();
    for (int s = BDIM / 2; s > 0; s >>= 1) {
        if (tid < s) red[tid] += red[tid + s];
        __syncthreads();
    }
    float z = red[0];
    if (is_s && lim == 0) z = 1.0f;
    if (tid < nit) w[tid] = sc[tid] / z;
    __syncthreads();

    const size_t row0 = is_s ? (size_t)MD + (size_t)ba * NC : (size_t)ba * NB;
    float* out = (is_s ? sv : f) + (size_t)ba * DCATD;
    for (int c = tid; c < DCATD; c += BDIM) {
        float a2 = 0.0f;
        for (int n = 0; n < nit; ++n)
            a2 += w[n] * X[(row0 + n) * DIN2D + c];
        out[c] = a2;
    }
}

// ---------------------------------------------------------------------------
// Kernel E: out[ba] = actb[ba] . (w_1 + T2[ba] + T3[ba]) + mbias
// ---------------------------------------------------------------------------
__global__ __launch_bounds__(BDIM)
void final_kernel(const float* __restrict__ actb, const float* __restrict__ w_1,
                  const float* __restrict__ T2, const float* __restrict__ T3,
                  const float* __restrict__ mbias, float* __restrict__ out)
{
    const int ba = blockIdx.x, tid = threadIdx.x;
    __shared__ float red[BDIM];
    const float* a = actb + (size_t)ba * DCATD;
    float acc = 0.0f;
    for (int c = tid; c < DCATD; c += BDIM)
        acc += a[c] * (w_1[c] + T2[(size_t)ba * DCATD + c] + T3[(size_t)ba * DCATD + c]);
    red[tid] = acc;
    __syncthreads();
    for (int s = BDIM / 2; s > 0; s >>= 1) {
        if (tid < s) red[tid] += red[tid + s];
        __syncthreads();
    }
    if (tid == 0) out[ba] = red[0] + mbias[0];
}

// ---------------------------------------------------------------------------
extern "C" void kernel_launch(void* const* d_in, const int* in_sizes, int n_in,
                              void* d_out, int out_size, void* d_ws, size_t ws_size,
                              hipStream_t stream) {
    const float* all_box      = (const float*)d_in[0];
    const float* all_box_pos  = (const float*)d_in[1];
    const float* chosen_box   = (const float*)d_in[2];
    const float* chosen_pos   = (const float*)d_in[3];
    const float* action_box   = (const float*)d_in[4];
    const float* action_pos   = (const float*)d_in[5];
    const float* action_a_box = (const float*)d_in[6];
    const float* action_a_pos = (const float*)d_in[7];
    const float* action_c_box = (const float*)d_in[8];
    const float* action_c_pos = (const float*)d_in[9];
    const float* W1    = (const float*)d_in[10];
    const float* W2    = (const float*)d_in[11];
    const float* b2    = (const float*)d_in[12];
    const float* W3    = (const float*)d_in[13];
    const float* w_1   = (const float*)d_in[14];
    const float* w_2   = (const float*)d_in[15];
    const float* w_3   = (const float*)d_in[16];
    const float* mbias = (const float*)d_in[17];
    const int*   box_num    = (const int*)d_in[18];
    const int*   chosen_num = (const int*)d_in[19];

    // Workspace layout (floats)
    float* ws    = (float*)d_ws;
    float* X     = ws;                                  // [3072, 8272] concat features
    float* Hbuf  = X    + (size_t)MTOT * DIN2D;         // [3072, 1024]
    float* sc    = Hbuf + (size_t)MTOT * H2D;           // [3072]
    float* actb  = sc   + MTOT;                         // [32, 4136]
    float* fbuf  = actb + (size_t)NBA * DCATD;          // [32, 4136]
    float* svbuf = fbuf + (size_t)NBA * DCATD;          // [32, 4136]
    float* T2    = svbuf + (size_t)NBA * DCATD;         // [32, 4136]
    float* T3    = T2   + (size_t)NBA * DCATD;          // [32, 4136]

    // 1) Build concatenated features. d-rows (0..2047) then s-rows (2048..3071).
    cat_rows<<<MD,  BDIM, 0, stream>>>(all_box,      all_box_pos,  W1, X,                       DIN2D, 0);
    cat_rows<<<MD,  BDIM, 0, stream>>>(action_a_box, action_a_pos, W1, X,                       DIN2D, DCATD);
    cat_rows<<<MS,  BDIM, 0, stream>>>(chosen_box,   chosen_pos,   W1, X + (size_t)MD * DIN2D,  DIN2D, 0);
    cat_rows<<<MS,  BDIM, 0, stream>>>(action_c_box, action_c_pos, W1, X + (size_t)MD * DIN2D,  DIN2D, DCATD);
    cat_rows<<<NBA, BDIM, 0, stream>>>(action_box,   action_pos,   W1, actb,                    DCATD, 0);

    // 2) H = tanh(X @ W2^T + b2)   -- the 52 GFLOP WMMA GEMM (TDM-fed)
    {
        dim3 g((MTOT + MT - 1) / MT, (H2D + 255) / 256);
        gemm_wmma<<<g, BDIM, 0, stream>>>(X, DIN2D, W2, DIN2D, b2, 1,
                                          Hbuf, H2D, MTOT, H2D, DIN2D);
    }

    // 3) score = exp(H @ W3^T)
    score_kernel<<<MTOT, BDIM, 0, stream>>>(Hbuf, W3, sc);

    // 4) normalize + pool -> f, sv
    pool_kernel<<<2 * NBA, BDIM, 0, stream>>>(sc, sc + MD, X, box_num, chosen_num,
                                              fbuf, svbuf);

    // 5) T2 = F @ w_2^T, T3 = SV @ w_3^T  (weights streamed once)
    {
        dim3 g((NBA + MT - 1) / MT, (DCATD + 255) / 256);
        gemm_wmma<<<g, BDIM, 0, stream>>>(fbuf,  DCATD, w_2, DCATD, nullptr, 0,
                                          T2, DCATD, NBA, DCATD, DCATD);
        gemm_wmma<<<g, BDIM, 0, stream>>>(svbuf, DCATD, w_3, DCATD, nullptr, 0,
                                          T3, DCATD, NBA, DCATD, DCATD);
    }

    // 6) out[ba] = actb . (w_1 + T2 + T3) + mbias
    final_kernel<<<NBA, BDIM, 0, stream>>>(actb, w_1, T2, T3, mbias, (float*)d_out);
}